// DifferentiableArray_12283606468146
// MI455X (gfx1250) — compile-verified
//
#include <hip/hip_runtime.h>

#define NN 65536   // array_size
#define BB 2048    // batch
#define DD 64      // embedding dim

typedef __attribute__((ext_vector_type(2))) float v2f;
typedef __attribute__((ext_vector_type(4))) float f4;
typedef __attribute__((ext_vector_type(8))) float v8f;

// ---------------------------------------------------------------------------
// Kernel 1: s[n] = dot(PE[n,:], w)  via V_WMMA_F32_16X16X4_F32.
// Trick: fold w into the A operand (A[m,k] = PE[row,k]*w[k]) and use B = 1s,
// which is layout-invariant; D[m,*] = s[row] replicated across all 16 columns.
// A layout (32-bit A 16x4): lanes 0-15 hold K=0,1 in vgpr0,1; lanes 16-31 K=2,3.
// C/D layout: lane<16 vgpr r -> (M=r, N=lane); lane>=16 -> (M=r+8, N=lane-16).
// Each wave produces 16 rows of s; 16 WMMAs accumulate over K=64.
// ---------------------------------------------------------------------------
__global__ void k_compute_s(const float* __restrict__ pe,   // [N, D]
                            const float* __restrict__ w,    // [D]
                            float* __restrict__ s) {        // [N]
  const int lane = threadIdx.x & 31;
  const int wave = (int)((blockIdx.x * blockDim.x + threadIdx.x) >> 5);
  const int n0 = wave * 16;          // 16 rows of s per wave
  const int m  = lane & 15;          // row within tile
  const int kh = (lane >> 4) * 2;    // K sub-offset for this lane half

  const float* arow = pe + (size_t)(n0 + m) * DD;
  v8f c = {};
  v2f one; one.x = 1.0f; one.y = 1.0f;
#pragma unroll
  for (int d0 = 0; d0 < DD; d0 += 4) {
    v2f a;
    a.x = arow[d0 + kh]     * w[d0 + kh];
    a.y = arow[d0 + kh + 1] * w[d0 + kh + 1];
    c = __builtin_amdgcn_wmma_f32_16x16x4_f32(false, a, false, one,
                                              (short)0, c, false, false);
  }
  // D[m,n] identical for all n: lane 0 holds s[n0+0..7], lane 16 s[n0+8..15].
  if (m == 0) {
    float* out = s + n0 + (lane >> 4) * 8;
#pragma unroll
    for (int r = 0; r < 8; ++r) out[r] = c[r];
  }
}

// ---------------------------------------------------------------------------
// Kernel 2: global max/min of s (single block). Row softmax max is then
// m_b = q>=0 ? q*max(s) : q*min(s) with no per-row pass.
// ---------------------------------------------------------------------------
__global__ void k_minmax(const float* __restrict__ s, float* __restrict__ mm) {
  __shared__ float smx[1024];
  __shared__ float smn[1024];
  const int t = threadIdx.x;
  float mx = -3.402823466e38f, mn = 3.402823466e38f;
  for (int i = t; i < NN; i += 1024) {
    float v = s[i];
    mx = fmaxf(mx, v);
    mn = fminf(mn, v);
  }
  smx[t] = mx; smn[t] = mn;
  __syncthreads();
  for (int off = 512; off > 0; off >>= 1) {
    if (t < off) {
      smx[t] = fmaxf(smx[t], smx[t + off]);
      smn[t] = fminf(smn[t], smn[t + off]);
    }
    __syncthreads();
  }
  if (t == 0) { mm[0] = smx[0]; mm[1] = smn[0]; }
}

// ---------------------------------------------------------------------------
// Kernel 3: per-row softmax stats. s is L2-resident (256 KiB in 192 MB L2),
// so this pass costs no HBM traffic. rowinv = 1/denominator.
// ---------------------------------------------------------------------------
__global__ void k_rowstats(const float* __restrict__ s,
                           const float* __restrict__ ni,
                           const float* __restrict__ mm,
                           float* __restrict__ rowm,
                           float* __restrict__ rowinv) {
  __shared__ float red[256];
  const int b = blockIdx.x;
  const int t = threadIdx.x;
  const float q = ni[b];
  const float m = (q >= 0.0f) ? q * mm[0] : q * mm[1];
  const f4* s4 = (const f4*)s;
  float acc = 0.0f;
  for (int i = t; i < NN / 4; i += 256) {
    f4 v = s4[i];
    acc += __expf(q * v.x - m) + __expf(q * v.y - m) +
           __expf(q * v.z - m) + __expf(q * v.w - m);
  }
  red[t] = acc;
  __syncthreads();
  for (int off = 128; off > 0; off >>= 1) {
    if (t < off) red[t] += red[t + off];
    __syncthreads();
  }
  if (t == 0) { rowm[b] = m; rowinv[b] = 1.0f / red[0]; }
}

// ---------------------------------------------------------------------------
// Kernel 4: the bandwidth pass. For a 1024-wide chunk of row b:
//   sim = q*s ; att = exp(sim-m)*inv ; partial += att*values
// Streamed arrays (values/sim/att, 512 MiB each) use non-temporal hints so
// they don't evict s from L2. Deterministic per-block partial sums.
// ---------------------------------------------------------------------------
__global__ void k_main(const float* __restrict__ values,
                       const float* __restrict__ ni,
                       const float* __restrict__ s,
                       const float* __restrict__ rowm,
                       const float* __restrict__ rowinv,
                       float* __restrict__ att,
                       float* __restrict__ sim,
                       float* __restrict__ partial) {
  __shared__ float red[256];
  const int b = blockIdx.y;
  const int t = threadIdx.x;
  const int n4 = blockIdx.x * 256 + t;                 // float4 index in row
  const size_t row4 = (size_t)b * (NN / 4) + n4;
  const float q   = ni[b];
  const float m   = rowm[b];
  const float inv = rowinv[b];

  f4 sv = ((const f4*)s)[n4];                          // L2-resident
  f4 simv;
  simv.x = q * sv.x; simv.y = q * sv.y; simv.z = q * sv.z; simv.w = q * sv.w;
  f4 av;
  av.x = __expf(simv.x - m) * inv;
  av.y = __expf(simv.y - m) * inv;
  av.z = __expf(simv.z - m) * inv;
  av.w = __expf(simv.w - m) * inv;

  __builtin_nontemporal_store(simv, ((f4*)sim) + row4);
  __builtin_nontemporal_store(av,  ((f4*)att) + row4);
  f4 vv = __builtin_nontemporal_load(((const f4*)values) + row4);

  red[t] = av.x * vv.x + av.y * vv.y + av.z * vv.z + av.w * vv.w;
  __syncthreads();
  for (int off = 128; off > 0; off >>= 1) {
    if (t < off) red[t] += red[t + off];
    __syncthreads();
  }
  if (t == 0) partial[(size_t)b * gridDim.x + blockIdx.x] = red[0];
}

// ---------------------------------------------------------------------------
// Kernel 5: deterministic final reduction of the 64 partials per row.
// ---------------------------------------------------------------------------
__global__ void k_final(const float* __restrict__ partial,
                        float* __restrict__ out) {
  __shared__ float red[64];
  const int b = blockIdx.x;
  const int t = threadIdx.x;
  red[t] = partial[(size_t)b * 64 + t];
  __syncthreads();
  for (int off = 32; off > 0; off >>= 1) {
    if (t < off) red[t] += red[t + off];
    __syncthreads();
  }
  if (t == 0) out[b] = red[0];
}

extern "C" void kernel_launch(void* const* d_in, const int* in_sizes, int n_in,
                              void* d_out, int out_size, void* d_ws, size_t ws_size,
                              hipStream_t stream) {
  const float* values = (const float*)d_in[0];  // [B,N]
  const float* ni     = (const float*)d_in[1];  // [B,1]
  const float* pe     = (const float*)d_in[2];  // [N,D]
  const float* w      = (const float*)d_in[3];  // [D,1] -> flat [D]

  // Output tuple, flattened in return order: output | attention | similarities
  float* out_output = (float*)d_out;                       // [B]
  float* out_att    = out_output + BB;                     // [B*N]
  float* out_sim    = out_att + (size_t)BB * NN;           // [B*N]

  // Workspace layout (floats): s[N] | mm[2] (pad) | rowm[B] | rowinv[B] | partial[B*64]
  float* ws      = (float*)d_ws;
  float* s       = ws;
  float* mm      = ws + NN;
  float* rowm    = ws + NN + 8;
  float* rowinv  = rowm + BB;
  float* partial = rowinv + BB;

  // 1) s = PE @ w via WMMA: 4096 waves (16 rows each), 8 waves/block.
  k_compute_s<<<NN / 16 / 8, 256, 0, stream>>>(pe, w, s);
  // 2) global max/min of s.
  k_minmax<<<1, 1024, 0, stream>>>(s, mm);
  // 3) per-row softmax max + denominator (L2-only traffic).
  k_rowstats<<<BB, 256, 0, stream>>>(s, ni, mm, rowm, rowinv);
  // 4) bandwidth pass: write sim+att, partial dot with values.
  k_main<<<dim3(NN / 1024, BB), 256, 0, stream>>>(values, ni, s, rowm, rowinv,
                                                  out_att, out_sim, partial);
  // 5) deterministic final reduce -> output[b].
  k_final<<<BB, 64, 0, stream>>>(partial, out_output);
}